// GINNet_43061342110477
// MI455X (gfx1250) — compile-verified
//
#include <hip/hip_runtime.h>

// ---------------------------------------------------------------------------
// GIN forward on MI455X (gfx1250, wave32).
// GEMMs: V_WMMA_F32_16X16X4_F32, B operand register-resident (64 VGPRs/wave),
// A operand staged to LDS (double buffered) via one of three CDNA5 paths:
//   STAGE_ASYNC : global_load_async_to_lds_b128 (ASYNCcnt)
//   STAGE_TDM   : tensor_load_to_lds, D# built per ISA 8.3/8.4 (TENSORcnt)
//   STAGE_XFORM : sync load -> relu(a*scale+shift) -> ds_store_b128 (fused BN)
// ---------------------------------------------------------------------------

#define N_NODES   50000
#define N_EDGES   600000
#define HID       128
#define N_CLASSES 10
#define N_LAYERS  4
#define N_GRAPHS  128
#define BN_EPS    1e-5f
#define TILES_PER_BLOCK 5   // 3125 row tiles = 625 blocks * 5

typedef __attribute__((ext_vector_type(2))) float        v2f;
typedef __attribute__((ext_vector_type(8))) float        v8f;
typedef __attribute__((ext_vector_type(4))) unsigned int v4u;
typedef __attribute__((ext_vector_type(8))) int          v8i;
typedef __attribute__((ext_vector_type(4))) int          v4i;

#if defined(__has_builtin)
#if __has_builtin(__builtin_amdgcn_tensor_load_to_lds) && \
    __has_builtin(__builtin_amdgcn_s_wait_tensorcnt)
#define HAVE_TDM 1
#endif
#endif
#ifndef HAVE_TDM
#define HAVE_TDM 0
#endif

enum { STAGE_ASYNC = 0, STAGE_TDM = 1, STAGE_XFORM = 2 };

// Low 32 bits of a flat shared-memory pointer == LDS byte offset on AMDGPU.
__device__ __forceinline__ uint32_t lds_off_u32(const void* p) {
  return (uint32_t)(uintptr_t)p;
}

__device__ __forceinline__ void wait_async_le2() {
  asm volatile("s_wait_asynccnt 0x2" ::: "memory");
}
__device__ __forceinline__ void wait_async_le0() {
  asm volatile("s_wait_asynccnt 0x0" ::: "memory");
}

// Async DMA staging: 512 x 16B chunks = 8KB tile, 2 instructions per thread.
__device__ __forceinline__ void stage_async(
    const float* __restrict__ A, int tile, float* __restrict__ buf, int tid)
{
  #pragma unroll
  for (int j = 0; j < 2; ++j) {
    const int c   = tid + 256 * j;
    const int row = c >> 5;
    const int kq  = (c & 31) << 2;
    const float* gsrc = A + ((size_t)(tile * 16 + row) << 7) + kq;
    const uint32_t lo = lds_off_u32(&buf[(row << 7) + kq]);
    asm volatile("global_load_async_to_lds_b128 %0, %1, off"
                 :: "v"(lo), "v"((unsigned long long)(uintptr_t)gsrc)
                 : "memory");
  }
}

#if HAVE_TDM
// One TDM descriptor moves the whole 16x128 fp32 tile (2D tensor, 4B elems).
// D# group0: count=1 | lds_addr | global_addr[56:0] | type=2.
// D# group1: data_size=4B, tensor_dim0=128, tensor_dim1=N, tile=128x16,
//            tensor_dim0_stride=128.  Issued by wave 0 only (EXEC ignored).
__device__ __forceinline__ void stage_tdm(
    const float* __restrict__ A, int tile, float* __restrict__ buf)
{
  const unsigned long long ga =
      (unsigned long long)(uintptr_t)(A + ((size_t)tile << 11));  // tile*2048
  v4u g0;
  g0.x = 1u;                                             // count=1, no gather
  g0.y = lds_off_u32(buf);                               // lds_addr
  g0.z = (unsigned int)ga;                               // global_addr[31:0]
  g0.w = (unsigned int)((ga >> 32) & 0x1FFFFFFu)         // global_addr[56:32]
       | (2u << 30);                                     // type=2 ("image")
  v8i g1;
  g1[0] = (int)(2u << 16);              // wg_mask=0, data_size=2 (4 bytes)
  g1[1] = (int)(128u << 16);            // tensor_dim0[15:0]=128 at [63:48]
  g1[2] = (int)((unsigned)N_NODES << 16); // tensor_dim1[15:0] at [95:80]
  g1[3] = (int)(128u << 16);            // tensor_dim1 hi=0 | tile_dim0=128
  g1[4] = 16;                           // tile_dim1=16, tile_dim2=0
  g1[5] = 128;                          // tensor_dim0_stride = 128 elems
  g1[6] = 0;                            // stride hi | tensor_dim1_stride lo
  g1[7] = 0;
  const v4i gz4 = {0, 0, 0, 0};                   // 2D: groups 2/3 unused
  const v8i gz8 = {0, 0, 0, 0, 0, 0, 0, 0};       // 6-arg form: extra group
  __builtin_amdgcn_tensor_load_to_lds(g0, g1, gz4, gz4, gz8, 0);
}
#endif

// Sync staging with fused BN+ReLU transform -> ds_store_b128.
__device__ __forceinline__ void stage_xform(
    const float* __restrict__ A, int tile, float* __restrict__ buf,
    const float* __restrict__ a_scale, const float* __restrict__ a_shift,
    int tid)
{
  #pragma unroll
  for (int j = 0; j < 2; ++j) {
    const int c   = tid + 256 * j;
    const int row = c >> 5;
    const int kq  = (c & 31) << 2;
    const float* gsrc = A + ((size_t)(tile * 16 + row) << 7) + kq;
    float4 v        = *(const float4*)gsrc;
    const float4 sc = *(const float4*)&a_scale[kq];
    const float4 sh = *(const float4*)&a_shift[kq];
    v.x = fmaxf(v.x * sc.x + sh.x, 0.0f);
    v.y = fmaxf(v.y * sc.y + sh.y, 0.0f);
    v.z = fmaxf(v.z * sc.z + sh.z, 0.0f);
    v.w = fmaxf(v.w * sc.w + sh.w, 0.0f);
    *(float4*)&buf[(row << 7) + kq] = v;                 // ds_store_b128
  }
}

// ---------------------------------------------------------------------------
// C[row, col] = stage(A)[row, :] @ W[:, col] + bias[col]
// Block = 256 threads (8 waves); wave w owns columns [16w, 16w+16).
// Block processes TILES_PER_BLOCK consecutive 16-row tiles, double buffered.
// VGPR layouts per CDNA5 ISA 7.12.2 (32-bit, 16x16x4):
//   A: lane l holds A[l&15][koff..koff+1], koff = 2*(l>>4)
//   B: lane l holds B[koff][l&15], B[koff+1][l&15]
//   C: vgpr i, lane l -> C[i + 8*(l>>4)][l&15]
// ---------------------------------------------------------------------------
template <int MODE>
__global__ __launch_bounds__(256) void gin_gemm_wmma(
    const float* __restrict__ A, const float* __restrict__ W,
    const float* __restrict__ bias,
    const float* __restrict__ a_scale, const float* __restrict__ a_shift,
    float* __restrict__ C)
{
  __shared__ float Abuf[2][16 * HID];   // 2 x 8KB ping-pong

  const int tid   = threadIdx.x;
  const int wave  = tid >> 5;
  const int lane  = tid & 31;
  const int m     = lane & 15;
  const int koff  = (lane >> 4) << 1;   // 0 or 2
  const int col   = (wave << 4) + m;
  const int tile0 = blockIdx.x * TILES_PER_BLOCK;

  // Register-resident B fragments: this wave's 16-column strip of W.
  v2f bfrag[32];
  #pragma unroll
  for (int kk = 0; kk < 32; ++kk) {
    const int k0 = (kk << 2) + koff;
    bfrag[kk].x = W[((size_t)k0 << 7) + col];
    bfrag[kk].y = W[((size_t)(k0 + 1) << 7) + col];
  }
  const float bb = bias[col];

  // Prologue stage of tile 0.
  if (MODE == STAGE_XFORM) {
    stage_xform(A, tile0, Abuf[0], a_scale, a_shift, tid);
#if HAVE_TDM
  } else if (MODE == STAGE_TDM) {
    if (wave == 0) stage_tdm(A, tile0, Abuf[0]);
#endif
  } else {
    stage_async(A, tile0, Abuf[0], tid);
  }

  for (int tt = 0; tt < TILES_PER_BLOCK; ++tt) {
    const int tile = tile0 + tt;
    const bool more = (tt + 1 < TILES_PER_BLOCK);

    if (MODE == STAGE_XFORM) {
      if (more) stage_xform(A, tile + 1, Abuf[(tt + 1) & 1],
                            a_scale, a_shift, tid);
#if HAVE_TDM
    } else if (MODE == STAGE_TDM) {
      if (wave == 0) {
        if (more) stage_tdm(A, tile + 1, Abuf[(tt + 1) & 1]);
        if (more) __builtin_amdgcn_s_wait_tensorcnt(1);
        else      __builtin_amdgcn_s_wait_tensorcnt(0);
      }
#endif
    } else {
      if (more) stage_async(A, tile + 1, Abuf[(tt + 1) & 1], tid);
      if (more) wait_async_le2();
      else      wait_async_le0();
    }
    __syncthreads();

    // Preload all 32 A fragments (16 x ds_load_2addr_b64, fully pipelined),
    // then 32 back-to-back WMMAs (D->C accumulate chain needs no NOPs).
    const float* __restrict__ ab = &Abuf[tt & 1][m << 7];
    v2f afrag[32];
    #pragma unroll
    for (int kk = 0; kk < 32; ++kk)
      afrag[kk] = *(const v2f*)&ab[(kk << 2) + koff];

    v8f acc = {};
    #pragma unroll
    for (int kk = 0; kk < 32; ++kk)
      acc = __builtin_amdgcn_wmma_f32_16x16x4_f32(
          false, afrag[kk], false, bfrag[kk], (short)0, acc, false, false);

    const int rbase = (tile << 4) + ((lane >> 4) << 3);
    #pragma unroll
    for (int i = 0; i < 8; ++i)
      C[((size_t)(rbase + i) << 7) + col] = acc[i] + bb;

    __syncthreads();   // reads of Abuf[tt&1] done before it is restaged
  }
}

// z0 = (1 + eps[layer]) * h
__global__ __launch_bounds__(256) void gin_init_scaled(
    const float* __restrict__ h, const float* __restrict__ eps, int layer,
    float* __restrict__ z0)
{
  const float e = 1.0f + eps[layer];
  size_t idx = (size_t)blockIdx.x * blockDim.x + threadIdx.x;
  if (idx < (size_t)N_NODES * HID) z0[idx] = e * h[idx];
}

// z0[dst] += h[src]; one wave per edge, lanes cover the row in 4 coalesced
// chunks of 32 contiguous floats.
__global__ __launch_bounds__(256) void gin_edge_scatter(
    const float* __restrict__ h, const int* __restrict__ src,
    const int* __restrict__ dst, float* __restrict__ z0)
{
  const int e = blockIdx.x * 8 + (threadIdx.x >> 5);
  if (e >= N_EDGES) return;
  const int lane = threadIdx.x & 31;
  const float* __restrict__ hs = h  + (size_t)src[e] * HID;
  float*       __restrict__ zd = z0 + (size_t)dst[e] * HID;
  #pragma unroll
  for (int j = 0; j < 4; ++j) {
    const int f = lane + 32 * j;
    atomicAdd(&zd[f], hs[f]);
  }
}

// Per-feature column sum / sum-of-squares partials.
__global__ __launch_bounds__(128) void gin_col_stats(
    const float* __restrict__ z, float* __restrict__ stats /* [256] */)
{
  const int f = threadIdx.x;
  float s = 0.0f, s2 = 0.0f;
  for (int r = blockIdx.x; r < N_NODES; r += gridDim.x) {
    const float v = z[(size_t)r * HID + f];
    s  += v;
    s2 += v * v;
  }
  atomicAdd(&stats[f], s);
  atomicAdd(&stats[HID + f], s2);
}

__global__ __launch_bounds__(128) void gin_bn_finalize(
    const float* __restrict__ stats, const float* __restrict__ g,
    const float* __restrict__ b, float* __restrict__ scale,
    float* __restrict__ shift)
{
  const int f = threadIdx.x;
  const float invn = 1.0f / (float)N_NODES;
  const float mean = stats[f] * invn;
  const float var  = stats[HID + f] * invn - mean * mean;
  const float sc   = g[f] * rsqrtf(var + BN_EPS);
  scale[f] = sc;
  shift[f] = b[f] - mean * sc;
}

// h += relu(z2*scale+shift); pooled[batch[node]] += h_new  (fused)
__global__ __launch_bounds__(256) void gin_residual_pool(
    float* __restrict__ h, const float* __restrict__ z2,
    const float* __restrict__ scale, const float* __restrict__ shift,
    const int* __restrict__ batch, float* __restrict__ pooled)
{
  size_t idx = (size_t)blockIdx.x * blockDim.x + threadIdx.x;
  if (idx >= (size_t)N_NODES * HID) return;
  const int node = (int)(idx >> 7);
  const int f    = (int)(idx & (HID - 1));
  const float v  = fmaxf(z2[idx] * scale[f] + shift[f], 0.0f);
  const float hn = h[idx] + v;
  h[idx] = hn;
  atomicAdd(&pooled[(size_t)batch[node] * HID + f], hn);
}

// pooled[batch[node]][f] += h[node][f]  (embedding layer)
__global__ __launch_bounds__(256) void gin_pool_accum(
    const float* __restrict__ h, const int* __restrict__ batch,
    float* __restrict__ pooled)
{
  size_t idx = (size_t)blockIdx.x * blockDim.x + threadIdx.x;
  if (idx >= (size_t)N_NODES * HID) return;
  const int node = (int)(idx >> 7);
  const int f    = (int)(idx & (HID - 1));
  atomicAdd(&pooled[(size_t)batch[node] * HID + f], h[idx]);
}

__global__ __launch_bounds__(256) void gin_counts(
    const int* __restrict__ batch, float* __restrict__ cnt)
{
  const int n = blockIdx.x * blockDim.x + threadIdx.x;
  if (n < N_NODES) atomicAdd(&cnt[batch[n]], 1.0f);
}

__global__ __launch_bounds__(256) void gin_zero(float* __restrict__ p, int n)
{
  for (int i = blockIdx.x * blockDim.x + threadIdx.x; i < n;
       i += gridDim.x * blockDim.x)
    p[i] = 0.0f;
}

// score[g][c] = sum_i ( (pooled[i][g]/cnt[g]) . pred_w[i][:,c] + pred_b[i][c] )
__global__ __launch_bounds__(32) void gin_final_score(
    const float* __restrict__ pooled, const float* __restrict__ cnt,
    const float* __restrict__ pw, const float* __restrict__ pb,
    float* __restrict__ out)
{
  const int g = blockIdx.x;
  const int c = threadIdx.x;
  if (c >= N_CLASSES) return;
  const float inv = 1.0f / fmaxf(cnt[g], 1.0f);
  float acc = 0.0f;
  for (int i = 0; i <= N_LAYERS; ++i) {
    const float* __restrict__ pg = pooled + ((size_t)i * N_GRAPHS + g) * HID;
    const float* __restrict__ wi = pw + (size_t)i * HID * N_CLASSES;
    float dot = 0.0f;
    #pragma unroll 4
    for (int k = 0; k < HID; ++k) dot += pg[k] * wi[k * N_CLASSES + c];
    acc += dot * inv + pb[i * N_CLASSES + c];
  }
  out[g * N_CLASSES + c] = acc;
}

// ---------------------------------------------------------------------------
extern "C" void kernel_launch(void* const* d_in, const int* in_sizes, int n_in,
                              void* d_out, int out_size, void* d_ws, size_t ws_size,
                              hipStream_t stream)
{
  const float* x     = (const float*)d_in[0];
  const int*   ei    = (const int*)  d_in[1];   // [2, E] row-major
  const int*   batch = (const int*)  d_in[2];
  const float* emb_w = (const float*)d_in[3];
  const float* emb_b = (const float*)d_in[4];
  const float* eps   = (const float*)d_in[5];
  const float* w1    = (const float*)d_in[6];
  const float* b1    = (const float*)d_in[7];
  const float* bn1g  = (const float*)d_in[8];
  const float* bn1b  = (const float*)d_in[9];
  const float* w2    = (const float*)d_in[10];
  const float* b2    = (const float*)d_in[11];
  const float* bng   = (const float*)d_in[12];
  const float* bnb   = (const float*)d_in[13];
  const float* pw    = (const float*)d_in[14];
  const float* pb    = (const float*)d_in[15];
  float* out = (float*)d_out;

  const size_t NH = (size_t)N_NODES * HID;
  float* ws     = (float*)d_ws;
  float* h      = ws;                           // [N, HID]
  float* z0     = h  + NH;                      // agg buffer (reused as z2)
  float* z1     = z0 + NH;
  float* stats  = z1 + NH;                      // sum[128] | sumsq[128]
  float* scale  = stats + 2 * HID;
  float* shift  = scale + HID;
  float* pooled = shift + HID;                  // [5, G, HID]
  float* cnt    = pooled + (size_t)(N_LAYERS + 1) * N_GRAPHS * HID;
  float* z2     = z0;

  const int GEMM_BLOCKS = (N_NODES / 16) / TILES_PER_BLOCK;  // 625
  const int EW_BLOCKS   = (int)((NH + 255) / 256);           // 25000
  const int EDGE_BLOCKS = (N_EDGES + 7) / 8;                 // 75000
  const int POOL_FLOATS = (N_LAYERS + 1) * N_GRAPHS * HID + N_GRAPHS;

  gin_zero<<<256, 256, 0, stream>>>(pooled, POOL_FLOATS);
  gin_counts<<<(N_NODES + 255) / 256, 256, 0, stream>>>(batch, cnt);

  // h = x @ emb_w + emb_b  (async-to-LDS staging path)
  gin_gemm_wmma<STAGE_ASYNC><<<GEMM_BLOCKS, 256, 0, stream>>>(
      x, emb_w, emb_b, nullptr, nullptr, h);
  gin_pool_accum<<<EW_BLOCKS, 256, 0, stream>>>(h, batch, pooled);

  for (int i = 0; i < N_LAYERS; ++i) {
    // z0 = (1+eps)*h + segment_sum(h[src] -> dst)
    gin_init_scaled<<<EW_BLOCKS, 256, 0, stream>>>(h, eps, i, z0);
    gin_edge_scatter<<<EDGE_BLOCKS, 256, 0, stream>>>(h, ei, ei + N_EDGES, z0);
    // z1 = z0 @ w1[i] + b1[i]   (TDM staging path when available)
#if HAVE_TDM
    gin_gemm_wmma<STAGE_TDM><<<GEMM_BLOCKS, 256, 0, stream>>>(
        z0, w1 + (size_t)i * HID * HID, b1 + i * HID, nullptr, nullptr, z1);
#else
    gin_gemm_wmma<STAGE_ASYNC><<<GEMM_BLOCKS, 256, 0, stream>>>(
        z0, w1 + (size_t)i * HID * HID, b1 + i * HID, nullptr, nullptr, z1);
#endif
    // BN1 stats
    gin_zero<<<1, 256, 0, stream>>>(stats, 2 * HID);
    gin_col_stats<<<512, 128, 0, stream>>>(z1, stats);
    gin_bn_finalize<<<1, 128, 0, stream>>>(stats, bn1g + i * HID,
                                           bn1b + i * HID, scale, shift);
    // z2 = relu(bn1(z1)) @ w2[i] + b2[i]  (BN+ReLU fused into staging)
    gin_gemm_wmma<STAGE_XFORM><<<GEMM_BLOCKS, 256, 0, stream>>>(
        z1, w2 + (size_t)i * HID * HID, b2 + i * HID, scale, shift, z2);
    // BN2 stats
    gin_zero<<<1, 256, 0, stream>>>(stats, 2 * HID);
    gin_col_stats<<<512, 128, 0, stream>>>(z2, stats);
    gin_bn_finalize<<<1, 128, 0, stream>>>(stats, bng + i * HID,
                                           bnb + i * HID, scale, shift);
    // h += relu(bn2(z2)) fused with pooling
    gin_residual_pool<<<EW_BLOCKS, 256, 0, stream>>>(
        h, z2, scale, shift, batch,
        pooled + (size_t)(i + 1) * N_GRAPHS * HID);
  }

  gin_final_score<<<N_GRAPHS, 32, 0, stream>>>(pooled, cnt, pw, pb, out);
}